// HeadAttention_73443940761659
// MI455X (gfx1250) — compile-verified
//
#include <hip/hip_runtime.h>
#include <math.h>

// ---------------------------------------------------------------------------
// HeadAttention: B=8, T=2048, E=1024, H=64, f32, multiplicative causal mask.
// All matmuls via V_WMMA_F32_16X16X4_F32 (exact f32, matches f32 reference).
// ---------------------------------------------------------------------------

typedef __attribute__((ext_vector_type(2))) float v2f;
typedef __attribute__((ext_vector_type(8))) float v8f;

#define T_SEQ 2048
#define EMB   1024
#define HID   64
#define NB    8
#define MTOT  (NB * T_SEQ)            // 16384 rows of X flattened
#define NKT   (T_SEQ / 16)            // 128 key tiles

// workspace layout (floats)
#define QOFF   0
#define KOFF   ((size_t)MTOT * HID)                 // 1,048,576
#define VOFF   ((size_t)2 * MTOT * HID)
#define TSOFF  ((size_t)3 * MTOT * HID)             // V tile sums [B][128][H]
#define SUFOFF (TSOFF + (size_t)NB * NKT * HID)     // V suffix    [B][129][H]

__device__ __forceinline__ v8f wmma_f32(v2f a, v2f b, v8f c) {
  // D = A(16x4 f32) x B(4x16 f32) + C(16x16 f32)
  return __builtin_amdgcn_wmma_f32_16x16x4_f32(
      /*neg_a=*/false, a, /*neg_b=*/false, b,
      /*c_mod=*/(short)0, c, /*reuse_a=*/false, /*reuse_b=*/false);
}

// ---------------------------------------------------------------------------
// Kernel 1: QKV projection.  X[M,E] @ W[E,H] + b -> ws.  One wave per 16x16
// output tile; K-loop of 256 f32 WMMAs.
// grid = (M/16, 3 matrices), block = (32, 4 n-tiles)
// ---------------------------------------------------------------------------
__global__ __launch_bounds__(128) void qkv_kernel(
    const float* __restrict__ X,
    const float* __restrict__ Wq, const float* __restrict__ bq,
    const float* __restrict__ Wk, const float* __restrict__ bk,
    const float* __restrict__ Wv, const float* __restrict__ bv,
    float* __restrict__ ws) {
  const int lane = threadIdx.x;            // 0..31
  const int nt   = threadIdx.y;            // 0..3 -> 16-col tile of H
  const int mat  = blockIdx.y;             // 0:Q 1:K 2:V
  const int m0   = blockIdx.x * 16;
  const int n0   = nt * 16;
  const int ln   = lane & 15;
  const int of   = (lane >> 4) * 2;        // K sub-offset 0 or 2 (A/B layout)

  const float* W    = (mat == 0) ? Wq : (mat == 1) ? Wk : Wv;
  const float* bias = (mat == 0) ? bq : (mat == 1) ? bk : bv;
  float* out = ws + (size_t)mat * (size_t)MTOT * HID;

  const float* arow = X + (size_t)(m0 + ln) * EMB + of;
  const float* bcol = W + (size_t)of * HID + n0 + ln;

  v8f c = {};
  #pragma unroll 4
  for (int k = 0; k < EMB; k += 4) {
    v2f a = *(const v2f*)(arow + k);       // A: lanes 0-15 K={k,k+1}, 16-31 K={k+2,k+3}
    v2f b;                                 // B: row k striped across lanes
    b.x = bcol[(size_t)k * HID];
    b.y = bcol[(size_t)(k + 1) * HID];
    c = wmma_f32(a, b, c);
  }

  const float bb = bias[n0 + ln];
  const int mrow = m0 + (lane >> 4) * 8;   // D layout: VGPR r -> M = r (+8 for hi half)
  #pragma unroll
  for (int r = 0; r < 8; ++r)
    out[(size_t)(mrow + r) * HID + n0 + ln] = c[r] + bb;
}

// ---------------------------------------------------------------------------
// Kernel 2a: per-16-row tile sums of V.   grid=(128,8), block=64 (h index)
// ---------------------------------------------------------------------------
__global__ void vtilesum_kernel(float* __restrict__ ws) {
  const int h = threadIdx.x;
  const int j = blockIdx.x;
  const int b = blockIdx.y;
  const float* V = ws + VOFF + (size_t)(b * T_SEQ + j * 16) * HID + h;
  float s = 0.f;
  #pragma unroll
  for (int i = 0; i < 16; ++i) s += V[(size_t)i * HID];
  ws[TSOFF + ((size_t)b * NKT + j) * HID + h] = s;
}

// ---------------------------------------------------------------------------
// Kernel 2b: suffix scan over tiles -> Suf[b][j][h] = sum_{k>=16j} V[b][k][h]
// grid=8, block=64
// ---------------------------------------------------------------------------
__global__ void vsuffix_kernel(float* __restrict__ ws) {
  const int h = threadIdx.x;
  const int b = blockIdx.x;
  const float* ts = ws + TSOFF + (size_t)b * NKT * HID;
  float* suf = ws + SUFOFF + (size_t)b * (NKT + 1) * HID;
  float acc = 0.f;
  suf[(size_t)NKT * HID + h] = 0.f;
  for (int j = NKT - 1; j >= 0; --j) {
    acc += ts[(size_t)j * HID + h];
    suf[(size_t)j * HID + h] = acc;
  }
}

// ---------------------------------------------------------------------------
// Kernel 3: flash attention, one wave per (batch, 16-row query tile).
// Multiplicative mask: masked scores are 0 and DO take part in softmax, so
// the whole future region contributes exp(-M) uniformly -> handled in closed
// form via the V suffix sums (halves the WMMA work vs scanning all 128 tiles).
// grid=(8, 32), block=(32,4)
// ---------------------------------------------------------------------------
__global__ __launch_bounds__(128) void attn_kernel(
    const float* __restrict__ ws, float* __restrict__ out) {
  __shared__ float plds[4][16][18];        // per-wave P transpose buffer (pad->8B align)

  const int lane = threadIdx.x;
  const int w    = threadIdx.y;
  const int b    = blockIdx.x;
  const int qt   = blockIdx.y * 4 + w;     // query tile 0..127 (wave-uniform)
  const int q0   = qt * 16;
  const int ln   = lane & 15;
  const int hi   = lane >> 4;              // 0/1
  const int of   = hi * 2;

  const float* Q = ws + QOFF;
  const float* K = ws + KOFF;
  const float* V = ws + VOFF;
  const float scale = 0.125f;              // 1/sqrt(64)

  // Q tile resident in A-layout registers: 16 K-steps of 4
  v2f qa[16];
  {
    const float* qrow = Q + (size_t)(b * T_SEQ + q0 + ln) * HID + of;
    #pragma unroll
    for (int kk = 0; kk < 16; ++kk) qa[kk] = *(const v2f*)(qrow + 4 * kk);
  }

  v8f acc0 = {}, acc1 = {}, acc2 = {}, acc3 = {};
  float rm[8], rs[8];
  #pragma unroll
  for (int r = 0; r < 8; ++r) { rm[r] = -INFINITY; rs[r] = 0.f; }

  for (int kt = 0; kt <= qt; ++kt) {
    const int k0 = kt * 16;

    // ---- S = Q . K^T (16 f32 WMMAs) ----
    v8f s = {};
    const float* krow = K + (size_t)(b * T_SEQ + k0 + ln) * HID + of;
    #pragma unroll
    for (int kk = 0; kk < 16; ++kk) {
      v2f kb = *(const v2f*)(krow + 4 * kk);   // B layout: row h lanes = key index
      s = wmma_f32(qa[kk], kb, s);
    }

    // ---- scale + multiplicative mask on the diagonal tile ----
    const int kg = k0 + ln;
    #pragma unroll
    for (int r = 0; r < 8; ++r) {
      const int qg = q0 + r + hi * 8;
      const float v = s[r] * scale;
      s[r] = (kg > qg) ? 0.f : v;              // masked score is literally 0
    }

    // ---- online softmax (row stats live across 16 lanes per half-wave) ----
    float corr[8];
    #pragma unroll
    for (int r = 0; r < 8; ++r) {
      float t = s[r];
      t = fmaxf(t, __shfl_xor(t, 1));
      t = fmaxf(t, __shfl_xor(t, 2));
      t = fmaxf(t, __shfl_xor(t, 4));
      t = fmaxf(t, __shfl_xor(t, 8));
      const float nm = fmaxf(rm[r], t);
      const float c  = __expf(rm[r] - nm);
      const float e  = __expf(s[r] - nm);
      s[r] = e;
      float rsum = e;
      rsum += __shfl_xor(rsum, 1);
      rsum += __shfl_xor(rsum, 2);
      rsum += __shfl_xor(rsum, 4);
      rsum += __shfl_xor(rsum, 8);
      rs[r] = rs[r] * c + rsum;
      rm[r] = nm;
      corr[r] = c;
    }
    #pragma unroll
    for (int r = 0; r < 8; ++r) {
      acc0[r] *= corr[r]; acc1[r] *= corr[r];
      acc2[r] *= corr[r]; acc3[r] *= corr[r];
    }

    // ---- transpose P via LDS: D layout -> A layout ----
    #pragma unroll
    for (int r = 0; r < 8; ++r) plds[w][r + hi * 8][ln] = s[r];
    asm volatile("s_wait_dscnt 0x0" ::: "memory");

    // ---- O += P . V (16 f32 WMMAs) ----
    const float* vb = V + (size_t)(b * T_SEQ + k0) * HID;
    #pragma unroll
    for (int j = 0; j < 4; ++j) {
      v2f pa = *(const v2f*)(&plds[w][ln][4 * j + of]);
      const float* vr0 = vb + (size_t)(4 * j + of) * HID + ln;
      const float* vr1 = vb + (size_t)(4 * j + of + 1) * HID + ln;
      v2f bb;
      bb.x = vr0[0];  bb.y = vr1[0];  acc0 = wmma_f32(pa, bb, acc0);
      bb.x = vr0[16]; bb.y = vr1[16]; acc1 = wmma_f32(pa, bb, acc1);
      bb.x = vr0[32]; bb.y = vr1[32]; acc2 = wmma_f32(pa, bb, acc2);
      bb.x = vr0[48]; bb.y = vr1[48]; acc3 = wmma_f32(pa, bb, acc3);
    }
  }

  // ---- closed-form future region: cf keys, each score 0 -> weight e^{-M} ----
  const int cf = T_SEQ - 16 * (qt + 1);
  if (cf > 0) {
    const float* suf = ws + SUFOFF + ((size_t)b * (NKT + 1) + (qt + 1)) * HID + ln;
    const float s0 = suf[0], s1 = suf[16], s2 = suf[32], s3 = suf[48];
    #pragma unroll
    for (int r = 0; r < 8; ++r) {
      const float nm = fmaxf(rm[r], 0.f);
      const float c  = __expf(rm[r] - nm);
      const float wz = __expf(-nm);
      rs[r] = rs[r] * c + wz * (float)cf;
      acc0[r] = acc0[r] * c + wz * s0;
      acc1[r] = acc1[r] * c + wz * s1;
      acc2[r] = acc2[r] * c + wz * s2;
      acc3[r] = acc3[r] * c + wz * s3;
    }
  }

  // ---- normalize & store ----
  float* orow = out + (size_t)(b * T_SEQ + q0 + hi * 8) * HID + ln;
  #pragma unroll
  for (int r = 0; r < 8; ++r) {
    const float inv = 1.f / rs[r];
    orow[(size_t)r * HID +  0] = acc0[r] * inv;
    orow[(size_t)r * HID + 16] = acc1[r] * inv;
    orow[(size_t)r * HID + 32] = acc2[r] * inv;
    orow[(size_t)r * HID + 48] = acc3[r] * inv;
  }
}

// ---------------------------------------------------------------------------
extern "C" void kernel_launch(void* const* d_in, const int* in_sizes, int n_in,
                              void* d_out, int out_size, void* d_ws, size_t ws_size,
                              hipStream_t stream) {
  const float* X  = (const float*)d_in[0];
  const float* Wq = (const float*)d_in[1];
  const float* bq = (const float*)d_in[2];
  const float* Wk = (const float*)d_in[3];
  const float* bk = (const float*)d_in[4];
  const float* Wv = (const float*)d_in[5];
  const float* bv = (const float*)d_in[6];
  float* ws  = (float*)d_ws;   // needs ~13.2 MB
  float* out = (float*)d_out;  // [B, T, H] f32

  qkv_kernel<<<dim3(MTOT / 16, 3), dim3(32, 4), 0, stream>>>(
      X, Wq, bq, Wk, bk, Wv, bv, ws);
  vtilesum_kernel<<<dim3(NKT, NB), 64, 0, stream>>>(ws);
  vsuffix_kernel<<<NB, 64, 0, stream>>>(ws);
  attn_kernel<<<dim3(NB, NKT / 4), dim3(32, 4), 0, stream>>>(ws, out);
}